// ReadingLayerReLU_13563506721326
// MI455X (gfx1250) — compile-verified
//
#include <hip/hip_runtime.h>
#include <hip/hip_bf16.h>

// Problem constants (from reference): B=8, S=2048, I=256, H=256, M=1024
#define BB 8
#define SS 2048
#define II 256
#define HH 256
#define MM 1024

#define S_TILE 64          // rows of s per workgroup (4 x 16-row WMMA subtiles)
#define LDS_PAD 4          // row pad (floats) -> conflict-free ds_load_b64
#define LDS_W (II + LDS_PAD)

typedef __attribute__((ext_vector_type(2))) float v2f;
typedef __attribute__((ext_vector_type(8))) float v8f;
typedef __attribute__((ext_vector_type(4))) float v4f;

// Fused: key = relu(x @ W^T) ; val = key @ mem^T  (per batch)
// One workgroup: 64 s-rows, 8 waves. Each wave owns ALL FOUR 16-row s-subtiles
// so every global B fragment feeds 4 WMMAs. The LDS tile buffer is reused:
// x tile during phase 1, key tile during phase 2 (barrier-separated).
__global__ __launch_bounds__(256)
void reading_layer_fused(const float* __restrict__ x,
                         const float* __restrict__ mem,
                         const float* __restrict__ W,
                         float* __restrict__ key_out,   // [B,S,H]
                         float* __restrict__ val_out) { // [B,S,M]
    __shared__ float lds_t[S_TILE][LDS_W];  // x tile, then key tile (64 x 256)

    const int tid   = threadIdx.x;
    const int lane  = tid & 31;
    const int wave  = tid >> 5;      // 0..7
    const int lmod  = lane & 15;     // N index within a 16x16 tile
    const int lhalf = lane >> 4;     // 0/1 : K-half selector for fp32 A/B frags

    const int b  = blockIdx.x / (SS / S_TILE);
    const int st = blockIdx.x % (SS / S_TILE);
    const int s0 = st * S_TILE;

    // ---- stage x tile into LDS (coalesced float4) ----
    {
        const float* xsrc = x + ((size_t)b * SS + s0) * II;
        for (int i = tid; i < (S_TILE * II) / 4; i += 256) {
            int row = i >> 6;              // II/4 = 64 vec4 per row
            int col = (i & 63) * 4;
            v4f v = *(const v4f*)(xsrc + (size_t)row * II + col);
            *(v4f*)&lds_t[row][col] = v;   // (row*LDS_W + col)*4 is 16B aligned
        }
    }
    __syncthreads();

    // ---- Phase 1: key tile = relu(x_tile @ W^T) ----
    // wave owns h columns [wave*32, wave*32+31] for all 64 s rows.
    const int h0 = wave * 32;
    v8f kacc[4][2] = {};               // [s-subtile][h-tile] : 64 VGPRs

#pragma unroll 4
    for (int k0 = 0; k0 < II; k0 += 4) {
        // A fragments (16x4 fp32): VGPR j holds K = k0 + j + 2*lhalf
        v2f a[4];
#pragma unroll
        for (int si = 0; si < 4; ++si)
            a[si] = *(const v2f*)&lds_t[16 * si + lmod][k0 + 2 * lhalf];
#pragma unroll
        for (int t = 0; t < 2; ++t) {
            const int h = h0 + t * 16 + lmod;
            // B fragment (4x16 fp32): B[k][n] = W[h=n][k]
            v2f bf = *(const v2f*)(W + (size_t)h * II + k0 + 2 * lhalf);
#pragma unroll
            for (int si = 0; si < 4; ++si)
                kacc[si][t] = __builtin_amdgcn_wmma_f32_16x16x4_f32(
                    false, a[si], false, bf, (short)0, kacc[si][t], false, false);
        }
    }

    __syncthreads();   // all waves done reading x tile; safe to overwrite

    // ReLU + write key to global and LDS (C/D layout: VGPR r -> M=r+8*lhalf)
#pragma unroll
    for (int si = 0; si < 4; ++si) {
#pragma unroll
        for (int t = 0; t < 2; ++t) {
            const int col = h0 + t * 16 + lmod;
#pragma unroll
            for (int r = 0; r < 8; ++r) {
                float v = fmaxf(kacc[si][t][r], 0.0f);
                const int row = 16 * si + r + 8 * lhalf;
                lds_t[row][col] = v;
                key_out[((size_t)b * SS + s0 + row) * HH + col] = v;
            }
        }
    }
    __syncthreads();

    // ---- Phase 2: val tile = key_tile @ mem[b]^T ----
    // wave owns m columns [wave*128, wave*128+127], in 2 chunks of 64.
    const float* memb = mem + (size_t)b * MM * II;
    const int m0w = wave * 128;

#pragma unroll 1
    for (int chunk = 0; chunk < 2; ++chunk) {
        const int mbase = m0w + chunk * 64;

        // prefetch next chunk's mem rows into cache
        if (chunk < 1) {
            __builtin_prefetch(memb + (size_t)(mbase + 64 + lmod * 4) * II, 0, 3);
        }

        v8f vacc[4][4] = {};           // [s-subtile][m-tile] : 128 VGPRs
#pragma unroll 2
        for (int k0 = 0; k0 < HH; k0 += 4) {
            v2f a[4];
#pragma unroll
            for (int si = 0; si < 4; ++si)
                a[si] = *(const v2f*)&lds_t[16 * si + lmod][k0 + 2 * lhalf];
#pragma unroll
            for (int t = 0; t < 4; ++t) {
                const int m = mbase + t * 16 + lmod;
                // B[k][n] = mem[b][m=n][h=k]
                v2f bf = *(const v2f*)(memb + (size_t)m * II + k0 + 2 * lhalf);
#pragma unroll
                for (int si = 0; si < 4; ++si)
                    vacc[si][t] = __builtin_amdgcn_wmma_f32_16x16x4_f32(
                        false, a[si], false, bf, (short)0, vacc[si][t], false, false);
            }
        }

#pragma unroll
        for (int si = 0; si < 4; ++si) {
#pragma unroll
            for (int t = 0; t < 4; ++t) {
                const int m = mbase + t * 16 + lmod;
#pragma unroll
                for (int r = 0; r < 8; ++r) {
                    const int row = 16 * si + r + 8 * lhalf;
                    val_out[((size_t)b * SS + s0 + row) * MM + m] = vacc[si][t][r];
                }
            }
        }
    }
}

extern "C" void kernel_launch(void* const* d_in, const int* in_sizes, int n_in,
                              void* d_out, int out_size, void* d_ws, size_t ws_size,
                              hipStream_t stream) {
    const float* x   = (const float*)d_in[0];   // [B,S,I]
    const float* mem = (const float*)d_in[1];   // [B,M,H]
    const float* W   = (const float*)d_in[2];   // [H,I]

    float* key_out = (float*)d_out;                          // [B,S,H] first
    float* val_out = key_out + (size_t)BB * SS * HH;         // [B,S,M] second

    dim3 grid(BB * (SS / S_TILE));
    dim3 block(256);
    reading_layer_fused<<<grid, block, 0, stream>>>(x, mem, W, key_out, val_out);
}